// Loss_reonstruct_687194767410
// MI455X (gfx1250) — compile-verified
//
#include <hip/hip_runtime.h>
#include <stdint.h>

#define Wd 1280
#define Hd 384
#define Cd 3
#define Nd 8

#define TW 128            // input tile width (8 WMMA column chunks)
#define OTW 126           // valid output cols per tile
#define OTH 14            // valid output rows per tile

typedef float v2f __attribute__((ext_vector_type(2)));
typedef float v8f __attribute__((ext_vector_type(8)));
typedef unsigned int v4u __attribute__((ext_vector_type(4)));
typedef int v4i __attribute__((ext_vector_type(4)));
typedef int v8i __attribute__((ext_vector_type(8)));

#if defined(__HIP_DEVICE_COMPILE__) && __has_builtin(__builtin_amdgcn_tensor_load_to_lds)
#define HAVE_TDM 1
#else
#define HAVE_TDM 0
#endif

// Fused warp + SSIM tile kernel.
// Tile: 16 input rows x 128 input cols -> 14 x 126 valid SSIM outputs.
// x = warp(src, disp, sign), y = tgt. Accumulates sum of clipped SSIM loss.
__global__ __launch_bounds__(256) void ssim_warp_kernel(
    const float* __restrict__ src, const float* __restrict__ tgt,
    const float* __restrict__ disp, float sign, double* __restrict__ accum)
{
  __shared__ float xT[16][TW];       // warped source tile
  __shared__ float yT[16][TW];       // target tile (TDM destination)
  __shared__ float Tq[5][16][TW];    // vertical 3-row box sums of {x,y,xx,yy,xy}
  __shared__ float red[256];

  const int tid = threadIdx.x;
  const int wv = tid >> 5;           // wave id (wave32): also the column chunk
  const int lane = tid & 31;
  const int nc = blockIdx.z;         // n*C + c
  const int n = nc / Cd;
  const int row0 = blockIdx.y * OTH;
  const int col0 = blockIdx.x * OTW;

#if HAVE_TDM
  // Wave 0 issues one TDM 2D-tile DMA of the dense target tile into LDS,
  // overlapping with the gather-based warp done by all waves below.
  if (wv == 0) {
    uint32_t lds_off = (uint32_t)(uintptr_t)(&yT[0][0]);
    uint64_t ga = (uint64_t)(uintptr_t)(tgt + ((size_t)nc * Hd + row0) * Wd + col0);
    uint32_t td0 = (uint32_t)(Wd - col0);   // distance to tensor edge (x)
    uint32_t td1 = (uint32_t)(Hd - row0);   // distance to tensor edge (y)
    v4u g0;
    g0.x = 1u;                                         // count=1, no gather
    g0.y = lds_off;                                    // lds_addr [63:32]
    g0.z = (uint32_t)ga;                               // global_addr lo
    g0.w = ((uint32_t)(ga >> 32) & 0x01FFFFFFu)        // global_addr hi
           | (2u << 30);                               // type=2 (image)
    v8i g1;
    g1[0] = (int)(2u << 16);                           // data_size=4B, mask=0
    g1[1] = (int)((td0 & 0xFFFFu) << 16);              // tensor_dim0 lo16
    g1[2] = (int)((td0 >> 16) | ((td1 & 0xFFFFu) << 16));
    g1[3] = (int)((td1 >> 16) | ((uint32_t)TW << 16)); // tile_dim0 = 128
    g1[4] = 16;                                        // tile_dim1 = 16
    g1[5] = Wd;                                        // tensor_dim0_stride lo
    g1[6] = 0;
    g1[7] = 0;
    v4i gz = {0, 0, 0, 0};
#if defined(__clang_major__) && __clang_major__ >= 23
    v8i gz8 = {0, 0, 0, 0, 0, 0, 0, 0};
    __builtin_amdgcn_tensor_load_to_lds(g0, g1, gz, gz, gz8, 0);
#else
    __builtin_amdgcn_tensor_load_to_lds(g0, g1, gz, gz, 0);
#endif
  }
#endif

  // Warp phase: bilinear horizontal gather (L2-resident source).
#pragma unroll
  for (int p = tid; p < 16 * TW; p += 256) {
    int r = p >> 7;                  // TW == 128
    int c = p & (TW - 1);
    int row = row0 + r, col = col0 + c;
    float xv = 0.f;
    if (row < Hd && col < Wd) {
      float d = disp[((size_t)n * Hd + row) * Wd + col];
      float xs = (float)col + sign * d;
      float x0 = floorf(xs);
      float wf = xs - x0;
      int x0i = (int)fminf(fmaxf(x0, 0.f), (float)(Wd - 1));
      int x1i = (int)fminf(fmaxf(x0 + 1.f, 0.f), (float)(Wd - 1));
      const float* srow = src + ((size_t)nc * Hd + row) * Wd;
      xv = (1.f - wf) * srow[x0i] + wf * srow[x1i];
    }
    xT[r][c] = xv;
#if !HAVE_TDM
    float yv = 0.f;
    if (row < Hd && col < Wd)
      yv = tgt[((size_t)nc * Hd + row) * Wd + col];
    yT[r][c] = yv;
#endif
  }

#if HAVE_TDM
  if (wv == 0) __builtin_amdgcn_s_wait_tensorcnt(0);
#endif
  __syncthreads();

  // WMMA phase: vertical 3-row box sums T = Band(16x16) * Q(16x16) on the
  // matrix pipe, f32 exact. Band matrix built from lane ids (0/1 entries).
  // Each wave owns one 16-column chunk (ci == wave id): straight-line code,
  // no divergence, EXEC all-ones as required by WMMA.
  {
    const int half = lane >> 4;
    const int ln = lane & 15;
    const int colb = (wv << 4) + ln;
    v8f accx  = {0.f, 0.f, 0.f, 0.f, 0.f, 0.f, 0.f, 0.f};
    v8f accy  = accx, accxx = accx, accyy = accx, accxy = accx;
#pragma unroll
    for (int k = 0; k < 4; ++k) {
      // A(16x4) layout: lanes 0-15 rows M; vgpr0 = K {4k, 4k+2}, vgpr1 = +1
      int K0 = 4 * k + (half ? 2 : 0);
      int K1 = K0 + 1;
      v2f a;
      a.x = (K0 >= ln && K0 <= ln + 2) ? 1.f : 0.f;   // band: rows m..m+2
      a.y = (K1 >= ln && K1 <= ln + 2) ? 1.f : 0.f;
      float x0v = xT[K0][colb], x1v = xT[K1][colb];
      float y0v = yT[K0][colb], y1v = yT[K1][colb];
      v2f bx, by, bxx, byy, bxy;
      bx.x  = x0v;       bx.y  = x1v;
      by.x  = y0v;       by.y  = y1v;
      bxx.x = x0v * x0v; bxx.y = x1v * x1v;
      byy.x = y0v * y0v; byy.y = y1v * y1v;
      bxy.x = x0v * y0v; bxy.y = x1v * y1v;
      accx  = __builtin_amdgcn_wmma_f32_16x16x4_f32(false, a, false, bx,  (short)0, accx,  false, false);
      accy  = __builtin_amdgcn_wmma_f32_16x16x4_f32(false, a, false, by,  (short)0, accy,  false, false);
      accxx = __builtin_amdgcn_wmma_f32_16x16x4_f32(false, a, false, bxx, (short)0, accxx, false, false);
      accyy = __builtin_amdgcn_wmma_f32_16x16x4_f32(false, a, false, byy, (short)0, accyy, false, false);
      accxy = __builtin_amdgcn_wmma_f32_16x16x4_f32(false, a, false, bxy, (short)0, accxy, false, false);
    }
    // C/D layout: vgpr v, lanes 0-15 -> M=v, lanes 16-31 -> M=v+8
#pragma unroll
    for (int v = 0; v < 8; ++v) {
      int mrow = v + 8 * half;
      Tq[0][mrow][colb] = accx[v];
      Tq[1][mrow][colb] = accy[v];
      Tq[2][mrow][colb] = accxx[v];
      Tq[3][mrow][colb] = accyy[v];
      Tq[4][mrow][colb] = accxy[v];
    }
  }
  __syncthreads();

  // Final phase: horizontal 3-sum + SSIM formula + clip, accumulate.
  float lsum = 0.f;
#pragma unroll
  for (int p = tid; p < OTH * OTW; p += 256) {
    int m = p / OTW;
    int j = p - m * OTW;
    int grow = row0 + m, gcol = col0 + j;
    if (grow <= Hd - 3 && gcol <= Wd - 3) {
      float sx  = Tq[0][m][j] + Tq[0][m][j + 1] + Tq[0][m][j + 2];
      float sy  = Tq[1][m][j] + Tq[1][m][j + 1] + Tq[1][m][j + 2];
      float sxx = Tq[2][m][j] + Tq[2][m][j + 1] + Tq[2][m][j + 2];
      float syy = Tq[3][m][j] + Tq[3][m][j + 1] + Tq[3][m][j + 2];
      float sxy = Tq[4][m][j] + Tq[4][m][j + 1] + Tq[4][m][j + 2];
      const float inv9 = 1.f / 9.f;
      float mux = sx * inv9, muy = sy * inv9;
      float sigx = sxx * inv9 - mux * mux;
      float sigy = syy * inv9 - muy * muy;
      float sigxy = sxy * inv9 - mux * muy;
      const float C1 = 0.0001f, C2 = 0.0009f;
      float nn = (2.f * mux * muy + C1) * (2.f * sigxy + C2);
      float dd = (mux * mux + muy * muy + C1) * (sigx + sigy + C2);
      float v = (1.f - nn / dd) * 0.5f;
      lsum += fminf(fmaxf(v, 0.f), 1.f);
    }
  }
  red[tid] = lsum;
  __syncthreads();
  for (int s = 128; s > 0; s >>= 1) {
    if (tid < s) red[tid] += red[tid + s];
    __syncthreads();
  }
  if (tid == 0) atomicAdd(accum, (double)red[0]);
}

// Left-right disparity consistency term.
__global__ __launch_bounds__(256) void lr_kernel(
    const float* __restrict__ ld, const float* __restrict__ rd,
    double* __restrict__ accum)
{
  __shared__ float red[256];
  const int total = Nd * Hd * Wd;
  float s = 0.f;
  for (int p = blockIdx.x * blockDim.x + threadIdx.x; p < total;
       p += gridDim.x * blockDim.x) {
    int col = p % Wd;
    int nh = p / Wd;
    float d = ld[p];
    float xs = (float)col - d;
    float x0 = floorf(xs);
    float wf = xs - x0;
    int x0i = (int)fminf(fmaxf(x0, 0.f), (float)(Wd - 1));
    int x1i = (int)fminf(fmaxf(x0 + 1.f, 0.f), (float)(Wd - 1));
    const float* row = rd + (size_t)nh * Wd;
    float v = (1.f - wf) * row[x0i] + wf * row[x1i];
    s += fabsf(d - v);
  }
  red[threadIdx.x] = s;
  __syncthreads();
  for (int st = 128; st > 0; st >>= 1) {
    if (threadIdx.x < st) red[threadIdx.x] += red[threadIdx.x + st];
    __syncthreads();
  }
  if (threadIdx.x == 0) atomicAdd(accum + 1, (double)red[0]);
}

__global__ void zero_kernel(double* accum) {
  accum[0] = 0.0;
  accum[1] = 0.0;
}

__global__ void finalize_kernel(const double* __restrict__ accum,
                                float* __restrict__ out) {
  // ssim count = N*C*(H-2)*(W-2) = 24*382*1278 ; lr count = N*H*W
  out[0] = (float)(accum[0] / 11716704.0 + accum[1] / 3932160.0);
}

extern "C" void kernel_launch(void* const* d_in, const int* in_sizes, int n_in,
                              void* d_out, int out_size, void* d_ws, size_t ws_size,
                              hipStream_t stream) {
  (void)in_sizes; (void)n_in; (void)out_size; (void)ws_size;
  const float* left_data  = (const float*)d_in[0];
  const float* right_data = (const float*)d_in[1];
  const float* left_disp  = (const float*)d_in[2];
  const float* right_disp = (const float*)d_in[3];
  float* out = (float*)d_out;
  double* accum = (double*)d_ws;

  zero_kernel<<<1, 1, 0, stream>>>(accum);

  dim3 grid((Wd - 2 + OTW - 1) / OTW,   // 11 col tiles
            (Hd - 2 + OTH - 1) / OTH,   // 28 row tiles
            Nd * Cd);
  // right term: x = warp(left_data, right_disp, +1), y = right_data
  ssim_warp_kernel<<<grid, 256, 0, stream>>>(left_data, right_data, right_disp,
                                             1.0f, accum);
  // left term: x = warp(right_data, left_disp, -1), y = left_data
  ssim_warp_kernel<<<grid, 256, 0, stream>>>(right_data, left_data, left_disp,
                                             -1.0f, accum);
  lr_kernel<<<4096, 256, 0, stream>>>(left_disp, right_disp, accum);
  finalize_kernel<<<1, 1, 0, stream>>>(accum, out);
}